// GraphAttentionLayer_39479339384841
// MI455X (gfx1250) — compile-verified
//
#include <hip/hip_runtime.h>
#include <hip/hip_bf16.h>
#include <math.h>

// ---------------------------------------------------------------------------
// GAT layer for MI455X (gfx1250, wave32).
//   h = x @ W                        -> V_WMMA_F32_16X16X4_F32 (dense f32 WMMA)
//   per-edge gather/score/scatter    -> wave-per-edge, float2/lane, f32 atomics
//   relu_bt global maxima            -> uint-bit atomicMax reductions
// ---------------------------------------------------------------------------

typedef float v2f __attribute__((ext_vector_type(2)));
typedef float v8f __attribute__((ext_vector_type(8)));

#define F_IN  128
#define F_OUT 64

__device__ __forceinline__ float leaky01(float x) { return x > 0.0f ? x : 0.01f * x; }

__device__ __forceinline__ float wave_max(float v) {
#pragma unroll
  for (int m = 16; m >= 1; m >>= 1) v = fmaxf(v, __shfl_xor(v, m, 32));
  return v;
}

__device__ __forceinline__ float wave_sum(float v) {
#pragma unroll
  for (int m = 16; m >= 1; m >>= 1) v += __shfl_xor(v, m, 32);
  return v;
}

// ---------------------------------------------------------------------------
// Kernel 1: h[N,64] = x[N,128] @ W[128,64] via V_WMMA_F32_16X16X4_F32.
// One wave computes a full 16x64 output strip (4 col-tiles, 4 accumulators):
//  - A (x row-tile) is loaded once per k-step and reused 4x
//  - 4 independent WMMA accumulate chains interleave on the matrix pipe
// 8 waves/block -> 8 row-tiles per block.
//
// f32 WMMA operand layout (ISA 7.12.2):
//   A 16x4 : lanes 0-15 hold M=lane  (VGPR0=K0, VGPR1=K1)
//            lanes16-31 hold M=lane-16 (VGPR0=K2, VGPR1=K3)
//   B 4x16 : lanes 0-15 col=lane   (VGPR0=row0, VGPR1=row1)
//            lanes16-31 col=lane-16 (VGPR0=row2, VGPR1=row3)
//   C 16x16: lanes 0-15 col=lane, VGPR i = row i; lanes16-31 rows 8..15.
// ---------------------------------------------------------------------------
__global__ void __launch_bounds__(256)
gat_gemm_wmma_f32(const float* __restrict__ x, const float* __restrict__ W,
                  float* __restrict__ h, int N) {
  const int wave    = threadIdx.x >> 5;
  const int lane    = threadIdx.x & 31;
  const int laneLo  = lane & 15;
  const bool hiHalf = lane >= 16;

  const int rowTile = blockIdx.x * 8 + wave;
  const int row0 = rowTile * 16;
  if (row0 >= N) return;                 // wave-uniform; EXEC stays all-ones

  const float* __restrict__ xrow = x + (size_t)(row0 + laneLo) * F_IN;
  const float* __restrict__ wbase = W + laneLo;

  v8f c0 = {}, c1 = {}, c2 = {}, c3 = {};
#pragma unroll 4
  for (int k0 = 0; k0 < F_IN; k0 += 4) {
    const int ka = k0 + (hiHalf ? 2 : 0);
    v2f a;
    a.x = xrow[ka];
    a.y = xrow[ka + 1];
    const float* __restrict__ wr0 = wbase + (size_t)(ka + 0) * F_OUT;
    const float* __restrict__ wr1 = wbase + (size_t)(ka + 1) * F_OUT;
    v2f b0, b1, b2, b3;
    b0.x = wr0[0];  b0.y = wr1[0];
    b1.x = wr0[16]; b1.y = wr1[16];
    b2.x = wr0[32]; b2.y = wr1[32];
    b3.x = wr0[48]; b3.y = wr1[48];
    // (neg_a, A, neg_b, B, c_mod, C, reuse_a, reuse_b)
    c0 = __builtin_amdgcn_wmma_f32_16x16x4_f32(false, a, false, b0, (short)0, c0, false, false);
    c1 = __builtin_amdgcn_wmma_f32_16x16x4_f32(false, a, false, b1, (short)0, c1, false, false);
    c2 = __builtin_amdgcn_wmma_f32_16x16x4_f32(false, a, false, b2, (short)0, c2, false, false);
    c3 = __builtin_amdgcn_wmma_f32_16x16x4_f32(false, a, false, b3, (short)0, c3, false, false);
  }

  const int outRow = row0 + (hiHalf ? 8 : 0);
  float* __restrict__ hp = h + (size_t)outRow * F_OUT + laneLo;
  if (row0 + 16 <= N) {
    // Full tile (always the case when N % 16 == 0): unconditional stores.
#pragma unroll
    for (int i = 0; i < 8; ++i) {
      float* r = hp + (size_t)i * F_OUT;
      r[0]  = c0[i];
      r[16] = c1[i];
      r[32] = c2[i];
      r[48] = c3[i];
    }
  } else {
#pragma unroll
    for (int i = 0; i < 8; ++i) {
      if (outRow + i < N) {
        float* r = hp + (size_t)i * F_OUT;
        r[0]  = c0[i];
        r[16] = c1[i];
        r[32] = c2[i];
        r[48] = c3[i];
      }
    }
  }
}

// ---------------------------------------------------------------------------
// Kernel 2: t_agg = max|hs+hd|, t_diff = max|hs-hd| over all edges.
// Wave per edge; lane handles feature components 2*lane, 2*lane+1.
// Non-negative floats compare identically as uint bit patterns -> atomicMax.
// ---------------------------------------------------------------------------
__global__ void __launch_bounds__(256)
gat_edge_maxes(const float* __restrict__ h, const int* __restrict__ src,
               const int* __restrict__ dst, unsigned* __restrict__ t_bits,
               long long E) {
  const int wave = threadIdx.x >> 5;
  const int lane = threadIdx.x & 31;
  const long long e = (long long)blockIdx.x * 8 + wave;
  if (e >= E) return;

  const int s = src[e];
  const int d = dst[e];
  const v2f hs = *(const v2f*)(h + (size_t)s * F_OUT + 2 * lane);
  const v2f hd = *(const v2f*)(h + (size_t)d * F_OUT + 2 * lane);

  const float ma = fmaxf(fabsf(hs.x + hd.x), fabsf(hs.y + hd.y));
  const float md = fmaxf(fabsf(hs.x - hd.x), fabsf(hs.y - hd.y));
  const float wa = wave_max(ma);
  const float wd = wave_max(md);
  if (lane == 0) {
    atomicMax(&t_bits[0], __float_as_uint(wa));
    atomicMax(&t_bits[1], __float_as_uint(wd));
  }
}

// ---------------------------------------------------------------------------
// Kernel 3: score[e] = hs.a0 + hd.a1 + relu_bt(hs+hd).a2 + relu_bt(hs-hd).a3
// using t_agg/t_diff from kernel 2; also reduce t_score = max|score|.
// ---------------------------------------------------------------------------
__global__ void __launch_bounds__(256)
gat_edge_score(const float* __restrict__ h, const int* __restrict__ src,
               const int* __restrict__ dst, const float* __restrict__ a,
               const unsigned* __restrict__ t_bits, float* __restrict__ score,
               unsigned* __restrict__ t_score_bits, long long E) {
  const int wave = threadIdx.x >> 5;
  const int lane = threadIdx.x & 31;
  const long long e = (long long)blockIdx.x * 8 + wave;
  if (e >= E) return;

  const float t_agg  = __uint_as_float(t_bits[0]);
  const float t_diff = __uint_as_float(t_bits[1]);

  const int s = src[e];
  const int d = dst[e];
  const int f = 2 * lane;
  const v2f hs = *(const v2f*)(h + (size_t)s * F_OUT + f);
  const v2f hd = *(const v2f*)(h + (size_t)d * F_OUT + f);
  const v2f a0 = *(const v2f*)(a + 0 * F_OUT + f);
  const v2f a1 = *(const v2f*)(a + 1 * F_OUT + f);
  const v2f a2 = *(const v2f*)(a + 2 * F_OUT + f);
  const v2f a3 = *(const v2f*)(a + 3 * F_OUT + f);

  const float agx = fminf(leaky01(hs.x + hd.x), t_agg);
  const float agy = fminf(leaky01(hs.y + hd.y), t_agg);
  const float dfx = fminf(leaky01(hs.x - hd.x), t_diff);
  const float dfy = fminf(leaky01(hs.y - hd.y), t_diff);

  float p = hs.x * a0.x + hs.y * a0.y + hd.x * a1.x + hd.y * a1.y +
            agx * a2.x + agy * a2.y + dfx * a3.x + dfy * a3.y;
  p = wave_sum(p);
  if (lane == 0) {
    score[e] = p;
    atomicMax(t_score_bits, __float_as_uint(fabsf(p)));
  }
}

// ---------------------------------------------------------------------------
// Kernel 4: edge_e = exp(-relu_bt(score)); scatter-add into h_prime / rowsum.
// ---------------------------------------------------------------------------
__global__ void __launch_bounds__(256)
gat_edge_scatter(const float* __restrict__ h, const int* __restrict__ src,
                 const int* __restrict__ dst, const float* __restrict__ score,
                 const unsigned* __restrict__ t_score_bits,
                 float* __restrict__ h_prime, float* __restrict__ e_rowsum,
                 long long E) {
  const int wave = threadIdx.x >> 5;
  const int lane = threadIdx.x & 31;
  const long long e = (long long)blockIdx.x * 8 + wave;
  if (e >= E) return;

  const float t_score = __uint_as_float(*t_score_bits);
  const float sc = score[e];
  const float edge_e = __expf(-fminf(leaky01(sc), t_score));

  const int s = src[e];
  const int d = dst[e];
  const int f = 2 * lane;
  const v2f hd = *(const v2f*)(h + (size_t)d * F_OUT + f);

  float* hp = h_prime + (size_t)s * F_OUT + f;
  atomicAdd(hp + 0, edge_e * hd.x);
  atomicAdd(hp + 1, edge_e * hd.y);
  if (lane == 0) atomicAdd(&e_rowsum[s], edge_e);
}

// ---------------------------------------------------------------------------
// Kernel 5: out = h_prime / (e_rowsum + 1e-16)
// ---------------------------------------------------------------------------
__global__ void __launch_bounds__(256)
gat_finalize(const float* __restrict__ h_prime, const float* __restrict__ e_rowsum,
             float* __restrict__ out, long long total) {
  const long long i = (long long)blockIdx.x * blockDim.x + threadIdx.x;
  if (i >= total) return;
  out[i] = h_prime[i] / (e_rowsum[i >> 6] + 1e-16f);
}

// ---------------------------------------------------------------------------
extern "C" void kernel_launch(void* const* d_in, const int* in_sizes, int n_in,
                              void* d_out, int out_size, void* d_ws, size_t ws_size,
                              hipStream_t stream) {
  const float* x  = (const float*)d_in[0];
  const int*   ei = (const int*)d_in[1];   // [2, E] int32
  const float* W  = (const float*)d_in[2]; // [128, 64]
  const float* a  = (const float*)d_in[3]; // [1, 256]

  const int       N = in_sizes[0] / F_IN;
  const long long E = (long long)in_sizes[1] / 2;
  const int* src = ei;
  const int* dst = ei + E;

  // Workspace layout (floats): h | h_prime | e_rowsum | score | t_bits[4]
  float* h        = (float*)d_ws;
  float* h_prime  = h + (size_t)N * F_OUT;
  float* e_rowsum = h_prime + (size_t)N * F_OUT;
  float* score    = e_rowsum + N;
  unsigned* t_bits = (unsigned*)(score + E);   // [0]=t_agg [1]=t_diff [2]=t_score

  // Zero the accumulator / reduction region (h and score are fully overwritten).
  const size_t zero_bytes =
      ((size_t)N * F_OUT + (size_t)N) * sizeof(float);  // h_prime + e_rowsum
  hipMemsetAsync(h_prime, 0, zero_bytes, stream);
  hipMemsetAsync(t_bits, 0, 4 * sizeof(unsigned), stream);

  // K1: GEMM. 8 waves/block, each wave = one 16x64 output strip.
  const int rowTiles = (N + 15) / 16;
  const int gemmBlocks = (rowTiles + 7) / 8;
  gat_gemm_wmma_f32<<<gemmBlocks, 256, 0, stream>>>(x, W, h, N);

  // Edge kernels: wave per edge, 8 waves per block.
  const int edgeBlocks = (int)((E + 7) / 8);
  gat_edge_maxes<<<edgeBlocks, 256, 0, stream>>>(h, src, dst, t_bits, E);
  gat_edge_score<<<edgeBlocks, 256, 0, stream>>>(h, src, dst, a, t_bits, score,
                                                 &t_bits[2], E);
  gat_edge_scatter<<<edgeBlocks, 256, 0, stream>>>(h, src, dst, score,
                                                   &t_bits[2], h_prime,
                                                   e_rowsum, E);

  // K5: normalize.
  const long long total = (long long)N * F_OUT;
  const int finBlocks = (int)((total + 255) / 256);
  gat_finalize<<<finBlocks, 256, 0, stream>>>(h_prime, e_rowsum,
                                              (float*)d_out, total);
}